// KernelPrediction_73615739453871
// MI455X (gfx1250) — compile-verified
//
#include <hip/hip_runtime.h>

typedef float v2f __attribute__((ext_vector_type(2)));
typedef float v4f __attribute__((ext_vector_type(4)));
typedef float v8f __attribute__((ext_vector_type(8)));

#define NCH 128
#define CH_STRIDE (64 * 64 * 64) /* 262144 */
#define Z_STRIDE (64 * 64)       /* 4096 */
#define Y_STRIDE 64

// One wave (32 threads) handles a tile of 16 points.
// Lane L<16  : point L,     channels c with c%4 in {0,1}
// Lane L>=16 : point L-16,  channels c with c%4 in {2,3}
// This matches the V_WMMA_F32_16X16X4_F32 A-matrix VGPR layout directly.
//
// Cache policy: the 128 MiB filter bank is kept L2-resident (default RT
// loads); all touch-once streams (coords, img, buddy, outputs) use
// non-temporal hints so they do not evict the bank from the 192 MB L2.
__global__ __launch_bounds__(32)
void trilinear_wmma_kernel(const float* __restrict__ nc,
                           const float* __restrict__ img,
                           const float* __restrict__ bank,
                           const float* __restrict__ bud,
                           float* __restrict__ out_sf,
                           float* __restrict__ out_final,
                           float* __restrict__ out_center,
                           float* __restrict__ out_res)
{
    __shared__ float lds_img[16 * NCH];
    __shared__ float lds_bud[16 * NCH];
    __shared__ float lds_sf [16 * NCH];

    const int lane = threadIdx.x;   // 0..31
    const int hi   = lane >> 4;     // 0/1 -> channel phase (K=0,1 vs K=2,3)
    const int prow = lane & 15;     // point row within tile (WMMA M index)
    const long base = (long)blockIdx.x * 16;

    // ---- coalesced, non-temporal stage of img & buddy tiles into LDS ----
    {
        const v4f* gi = (const v4f*)(img + base * NCH);
        const v4f* gb = (const v4f*)(bud + base * NCH);
        v4f* li = (v4f*)lds_img;
        v4f* lb = (v4f*)lds_bud;
#pragma unroll
        for (int t = 0; t < 16; ++t) {
            li[t * 32 + lane] = __builtin_nontemporal_load(&gi[t * 32 + lane]);
            lb[t * 32 + lane] = __builtin_nontemporal_load(&gb[t * 32 + lane]);
        }
    }
    __syncthreads();

    // ---- per-lane point coordinates & axis setup (D=H=W=64) ----
    const long p = base + prow;
    const float cx = __builtin_nontemporal_load(&nc[p * 3 + 0]);  // -> W (x)
    const float cy = __builtin_nontemporal_load(&nc[p * 3 + 1]);  // -> H (y)
    const float cz = __builtin_nontemporal_load(&nc[p * 3 + 2]);  // -> D (z)

    float ix = fminf(fmaxf((cx + 1.0f) * 0.5f * 63.0f, 0.0f), 63.0f);
    float iy = fminf(fmaxf((cy + 1.0f) * 0.5f * 63.0f, 0.0f), 63.0f);
    float iz = fminf(fmaxf((cz + 1.0f) * 0.5f * 63.0f, 0.0f), 63.0f);
    float fx = floorf(ix), fy = floorf(iy), fz = floorf(iz);
    const float wx = ix - fx, wy = iy - fy, wz = iz - fz;
    const float mx = 1.0f - wx, my = 1.0f - wy, mz = 1.0f - wz;
    const int x0 = (int)fx, y0 = (int)fy, z0 = (int)fz;
    const int ox = (x0 < 63) ? 1 : 0;
    const int oy = (y0 < 63) ? Y_STRIDE : 0;
    const int oz = (z0 < 63) ? Z_STRIDE : 0;
    const float* bp0 = bank + (size_t)(z0 * Z_STRIDE + y0 * Y_STRIDE + x0);

    v8f acc_res = {0.f, 0.f, 0.f, 0.f, 0.f, 0.f, 0.f, 0.f};
    v8f acc_bud = {0.f, 0.f, 0.f, 0.f, 0.f, 0.f, 0.f, 0.f};
    const v2f ones = {1.0f, 1.0f};

    // ---- main loop: 32 chunks of K=4 channels; each lane samples 2 channels ----
    for (int k = 0; k < 32; ++k) {
        const int c0 = 4 * k + 2 * hi;
        v2f aP, aQ;
#pragma unroll
        for (int j = 0; j < 2; ++j) {
            const int c = c0 + j;
            const float* bp = bp0 + (size_t)c * CH_STRIDE;
            const float v000 = bp[0];
            const float v001 = bp[ox];
            const float v010 = bp[oy];
            const float v011 = bp[oy + ox];
            const float v100 = bp[oz];
            const float v101 = bp[oz + ox];
            const float v110 = bp[oz + oy];
            const float v111 = bp[oz + oy + ox];
            const float c00 = v000 * mx + v001 * wx;
            const float c01 = v010 * mx + v011 * wx;
            const float c10 = v100 * mx + v101 * wx;
            const float c11 = v110 * mx + v111 * wx;
            const float d0  = c00 * my + c01 * wy;
            const float d1  = c10 * my + c11 * wy;
            const float s   = d0 * mz + d1 * wz;
            lds_sf[prow * NCH + c] = s;
            const float vi = lds_img[prow * NCH + c];
            const float vb = lds_bud[prow * NCH + c];
            aP[j] = vi * s;   // residual  = img .* sampled
            aQ[j] = vi * vb;  // center    = img .* buddy
        }
        // Row-sum via WMMA with B = ones: D[m, n] += sum_k A[m, k]
        acc_res = __builtin_amdgcn_wmma_f32_16x16x4_f32(
            false, aP, false, ones, (short)0, acc_res, false, false);
        acc_bud = __builtin_amdgcn_wmma_f32_16x16x4_f32(
            false, aQ, false, ones, (short)0, acc_bud, false, false);
    }

    __syncthreads();

    // ---- coalesced, non-temporal writeback of sampled filters ----
    {
        v4f* go = (v4f*)(out_sf + base * NCH);
        const v4f* ls = (const v4f*)lds_sf;
#pragma unroll
        for (int t = 0; t < 16; ++t)
            __builtin_nontemporal_store(ls[t * 32 + lane], &go[t * 32 + lane]);
    }

    // ---- scalar outputs ----
    // C/D layout: VGPR r holds M=r (lanes 0-15) and M=r+8 (lanes 16-31);
    // all N columns are equal (B = ones), so lane 0 has rows 0..7 and
    // lane 16 has rows 8..15 directly in its 8 accumulator registers.
    const v8f tot = acc_res + acc_bud;
    if ((lane & 15) == 0) {
        const long q = base + hi * 8;
#pragma unroll
        for (int r = 0; r < 8; ++r) {
            __builtin_nontemporal_store(acc_bud[r], &out_center[q + r]);
            __builtin_nontemporal_store(acc_res[r], &out_res[q + r]);
            __builtin_nontemporal_store(tot[r],     &out_final[q + r]);
        }
    }
}

extern "C" void kernel_launch(void* const* d_in, const int* in_sizes, int n_in,
                              void* d_out, int out_size, void* d_ws, size_t ws_size,
                              hipStream_t stream) {
    (void)n_in; (void)out_size; (void)d_ws; (void)ws_size;
    const float* nc   = (const float*)d_in[0];  // (N, 3)
    const float* img  = (const float*)d_in[1];  // (N, C)
    const float* bank = (const float*)d_in[2];  // (1, C, 64, 64, 64)
    const float* bud  = (const float*)d_in[3];  // (N, C)

    const int N = in_sizes[0] / 3;              // 131072

    float* out       = (float*)d_out;
    float* out_sf    = out;                        // (N, C)
    float* out_final = out + (size_t)N * NCH;      // (N,)
    float* out_ctr   = out_final + N;              // (N,)
    float* out_res   = out_ctr + N;                // (N,)

    const int tiles = N / 16;                   // N is a multiple of 16
    trilinear_wmma_kernel<<<tiles, 32, 0, stream>>>(
        nc, img, bank, bud, out_sf, out_final, out_ctr, out_res);
}